// GenSP_81071802680035
// MI455X (gfx1250) — compile-verified
//
#include <hip/hip_runtime.h>

// ---------------------------------------------------------------------------
// SSN superpixel affinity (B=4, C=32, H=W=256, s=16, N_ITER=2, M=0 -> grid
// channels are identically zero and are skipped with no effect on results).
//
// Memory-bound problem: output abs_aff is 4*256*65536 f32 = 268MB (~11.5us
// write floor at 23.3 TB/s). Compute ~0.8 GFLOP -> free. We still route the
// two GEMM-shaped pieces through V_WMMA_F32_16X16X4_F32 (f32-exact):
//   dots: D[p][k] = sum_c px[p][c]*cent[k][c]   (256x16x32 per cell)
//   update: num[k][c] = sum_p aff[k][p]*px[p][c] (16x32x256 per cell)
// ---------------------------------------------------------------------------

typedef __attribute__((ext_vector_type(2))) float v2f;
typedef __attribute__((ext_vector_type(8))) float v8f;

#define B_   4
#define C_   32
#define W_   256
#define NH_  16          // cells per side
#define S_   256         // superpixels
#define N_   65536       // pixels per image
#define BIGF 1e16f
static const long long TOT_AFF = (long long)B_ * S_ * N_;   // 67108864

// ---------------- K1: initial centroids (16x16 average pool) ----------------
__global__ __launch_bounds__(256) void centroid_init(const float* __restrict__ x,
                                                     float* __restrict__ cent) {
    int s = blockIdx.x, b = blockIdx.y;
    int cy = s >> 4, cx = s & 15;
    int t = threadIdx.x;
    int c = t & 31, g = t >> 5;                 // 32 channels x 8 groups
    const float* base = x + (((size_t)b * C_ + c) << 16);
    float acc = 0.f;
    for (int pp = g * 32; pp < g * 32 + 32; ++pp) {
        int iy = pp >> 4, ix = pp & 15;
        acc += base[(size_t)(cy * 16 + iy) * W_ + (cx * 16 + ix)];
    }
    __shared__ float red[256];
    red[t] = acc;
    __syncthreads();
    if (g == 0) {
        float sum = 0.f;
        #pragma unroll
        for (int gg = 0; gg < 8; ++gg) sum += red[c + (gg << 5)];
        cent[((size_t)(b * S_) + s) * C_ + c] = sum * (1.0f / 256.0f);
    }
}

// ---------------- small utility kernels ----------------
__global__ void zero_f32(float* __restrict__ p, long long n) {
    long long i = (long long)blockIdx.x * blockDim.x + threadIdx.x;
    long long stride = (long long)gridDim.x * blockDim.x;
    for (; i < n; i += stride) p[i] = 0.f;
}

__global__ void zero_out_vec4(float* __restrict__ p, long long n) {
    long long n4 = n >> 2;
    long long i = (long long)blockIdx.x * blockDim.x + threadIdx.x;
    long long stride = (long long)gridDim.x * blockDim.x;
    float4 z = make_float4(0.f, 0.f, 0.f, 0.f);
    float4* p4 = (float4*)p;
    for (long long j = i; j < n4; j += stride) p4[j] = z;
    long long tail = n4 << 2;
    if (i < (n - tail)) p[tail + i] = 0.f;
}

__global__ void centroid_div(const float* __restrict__ num,
                             const float* __restrict__ den,
                             float* __restrict__ cent) {
    int idx = blockIdx.x * blockDim.x + threadIdx.x;   // 32768 = B*S*C
    if (idx < B_ * S_ * C_)
        cent[idx] = num[idx] / (den[idx >> 5] + 1e-16f);
}

// ---------------- K2/K4: affinity pass (shared body) ----------------
// One block per (cell, batch). 256 threads = 8 waves = 256 pixels of the cell.
template <bool UPDATE>
__global__ __launch_bounds__(256) void assign_kernel(const float* __restrict__ x,
                                                     const float* __restrict__ cent,
                                                     float* __restrict__ num,
                                                     float* __restrict__ den,
                                                     float* __restrict__ out,
                                                     long long out_size) {
    int scell = blockIdx.x, b = blockIdx.y;
    int cy = scell >> 4, cx = scell & 15;
    int t = threadIdx.x;
    int lane = t & 31, wave = t >> 5;
    bool hi = lane >= 16;
    int l16 = lane & 15;
    int koff0 = hi ? 2 : 0, koff1 = hi ? 3 : 1;

    __shared__ float pxL[256][33];   // pixel tile (padded)
    __shared__ float centL[16][33];  // 9 valid centroid rows, zero padded
    __shared__ float dotL[256][17];  // dot products, then affinities (padded)
    __shared__ float cnL[16];
    __shared__ int   nspL[16], nvalidL[16];

    // neighbor superpixel table (k = dy*3+dx ordering matches reference)
    if (t < 16) {
        int k = t;
        int dy = k / 3 - 1, dx = k % 3 - 1;
        int ny = cy + dy, nx = cx + dx;
        int v = (k < 9) && (ny >= 0) && (ny < NH_) && (nx >= 0) && (nx < NH_);
        nvalidL[k] = v;
        nspL[k] = v ? (ny * NH_ + nx) : 0;
    }
    __syncthreads();

    // stage pixel tile (32 channels) and own |px|^2
    int iy = t >> 4, ix = t & 15;
    int gy = cy * 16 + iy, gx = cx * 16 + ix;
    const float* px = x + ((size_t)b * C_) * N_ + (size_t)gy * W_ + gx;
    __builtin_prefetch(px, 0, 0);                  // global_prefetch_b8
    float pxn = 0.f;
    #pragma unroll
    for (int c = 0; c < C_; ++c) {
        float v = px[(size_t)c * N_];
        pxL[t][c] = v;
        pxn += v * v;
    }
    // stage the (up to) 9 neighbor centroids, zero-padded to 16 rows
    for (int idx = t; idx < 16 * 32; idx += 256) {
        int k = idx >> 5, c = idx & 31;
        centL[k][c] = nvalidL[k] ? cent[((size_t)(b * S_) + nspL[k]) * C_ + c] : 0.f;
    }
    __syncthreads();
    if (t < 16) {
        float s2 = 0.f;
        #pragma unroll
        for (int c = 0; c < C_; ++c) { float v = centL[t][c]; s2 += v * v; }
        cnL[t] = s2;
    }
    __syncthreads();

    // ---- WMMA: D[p][k] = sum_c px[p][c] * cent[k][c], f32 16x16x4, K=32 ----
    // A 16x4 layout: lane<16 -> K{0,1}, lane>=16 -> K{2,3}; M = lane&15.
    for (int h = 0; h < 2; ++h) {
        int prow = wave * 32 + h * 16 + l16;       // A-row pixel for this lane
        v8f acc = {0.f, 0.f, 0.f, 0.f, 0.f, 0.f, 0.f, 0.f};
        #pragma unroll
        for (int kk = 0; kk < C_; kk += 4) {
            v2f a, bm;
            a[0]  = pxL[prow][kk + koff0];
            a[1]  = pxL[prow][kk + koff1];
            bm[0] = centL[l16][kk + koff0];        // B[k][n] = cent[n][k]
            bm[1] = centL[l16][kk + koff1];
            acc = __builtin_amdgcn_wmma_f32_16x16x4_f32(
                false, a, false, bm, (short)0, acc, false, false);
        }
        #pragma unroll
        for (int r = 0; r < 8; ++r) {              // D: VGPR r -> M = r (+8 hi)
            int m = hi ? r + 8 : r;
            dotL[wave * 32 + h * 16 + m][l16] = acc[r];
        }
    }
    __syncthreads();

    // ---- per-pixel masked softmax over the 9 neighbors ----
    float dist[9];
    float dmin = BIGF;
    #pragma unroll
    for (int k = 0; k < 9; ++k) {
        float d = nvalidL[k] ? (pxn - 2.f * dotL[t][k] + cnL[k]) : BIGF;
        dist[k] = d;
        dmin = fminf(dmin, d);
    }
    float e[9], esum = 0.f;
    #pragma unroll
    for (int k = 0; k < 9; ++k) {
        float ek = nvalidL[k] ? expf(dmin - dist[k]) : 0.f;
        e[k] = ek;
        esum += ek;
    }
    float inv = 1.f / esum;

    if (UPDATE) {
        // write affinities back into dotL (zero-padded rows k=9..15)
        #pragma unroll
        for (int k = 0; k < 16; ++k)
            dotL[t][k] = (k < 9 && nvalidL[k]) ? e[k] * inv : 0.f;
        __syncthreads();

        // den: 9 column sums + global atomic
        if (t < 9 && nvalidL[t]) {
            float sden = 0.f;
            for (int p = 0; p < 256; ++p) sden += dotL[p][t];
            atomicAdd(&den[b * S_ + nspL[t]], sden);
        }
        // ---- WMMA: num[k][c] = sum_p aff[k][p] * px[p][c], K = 256 pixels ----
        if (wave == 0) {
            for (int ct = 0; ct < 2; ++ct) {       // two 16-channel tiles
                v8f acc = {0.f, 0.f, 0.f, 0.f, 0.f, 0.f, 0.f, 0.f};
                for (int kk = 0; kk < 256; kk += 4) {
                    v2f a, bm;
                    a[0]  = dotL[kk + koff0][l16]; // A[k][p] = aff
                    a[1]  = dotL[kk + koff1][l16];
                    bm[0] = pxL[kk + koff0][ct * 16 + l16];
                    bm[1] = pxL[kk + koff1][ct * 16 + l16];
                    acc = __builtin_amdgcn_wmma_f32_16x16x4_f32(
                        false, a, false, bm, (short)0, acc, false, false);
                }
                #pragma unroll
                for (int r = 0; r < 8; ++r) {
                    int k = hi ? r + 8 : r;
                    if (k < 9 && nvalidL[k])
                        atomicAdd(&num[((size_t)(b * S_) + nspL[k]) * C_ + ct * 16 + l16],
                                  acc[r]);
                }
            }
        }
    } else {
        // scatter abs_aff[b][s][n]; n is image-linear pixel index
        size_t pixlin = (size_t)gy * W_ + gx;
        #pragma unroll
        for (int k = 0; k < 9; ++k)
            if (nvalidL[k])
                out[((size_t)(b * S_) + nspL[k]) * N_ + pixlin] = e[k] * inv;
        // trailing scalar output S (second tuple element), if present
        if (b == 0 && scell == 0 && t == 0 && out_size > TOT_AFF)
            out[TOT_AFF] = (float)S_;
    }
}

// ---------------------------------------------------------------------------
extern "C" void kernel_launch(void* const* d_in, const int* in_sizes, int n_in,
                              void* d_out, int out_size, void* d_ws, size_t ws_size,
                              hipStream_t stream) {
    const float* x = (const float*)d_in[0];
    float* out = (float*)d_out;
    float* ws = (float*)d_ws;

    // ws layout (floats): cent0[32768] | num[32768] | den[1024] | cent1[32768]
    float* cent0 = ws;
    float* num   = ws + 32768;
    float* den   = ws + 65536;
    float* cent1 = ws + 66560;

    dim3 gridBS(S_, B_);

    centroid_init<<<gridBS, 256, 0, stream>>>(x, cent0);
    zero_f32<<<132, 256, 0, stream>>>(num, 33792);                // num+den contiguous
    assign_kernel<true><<<gridBS, 256, 0, stream>>>(x, cent0, num, den, nullptr,
                                                    (long long)out_size);
    centroid_div<<<128, 256, 0, stream>>>(num, den, cent1);
    zero_out_vec4<<<2048, 256, 0, stream>>>(out, (long long)out_size);
    assign_kernel<false><<<gridBS, 256, 0, stream>>>(x, cent1, num, den, out,
                                                     (long long)out_size);
}